// model_LSTM_1_51092930953628
// MI455X (gfx1250) — compile-verified
//
#include <hip/hip_runtime.h>

// ---------------------------------------------------------------------------
// LSTM(1 -> 21) scan over 16384 steps + Linear(21 -> 1).
// Latency-bound serial recurrence: single wave32, weights in VGPRs,
// h broadcast via v_readlane, x column streamed into LDS by the CDNA5
// Tensor Data Mover one tile ahead of the scan, activations via v_tanh_f32.
// ---------------------------------------------------------------------------

#define SEQ_LEN 16384
#define N_FEAT  3
#define HID     21
#define G4      (4 * HID)        // 84 gate rows
#define TS      128              // timesteps per tile
#define NTILES  (SEQ_LEN / TS)   // 128

typedef unsigned int v4u __attribute__((ext_vector_type(4)));
typedef int          v8i __attribute__((ext_vector_type(8)));
typedef int          v4i __attribute__((ext_vector_type(4)));

#if __has_builtin(__builtin_amdgcn_tensor_load_to_lds) && \
    __has_builtin(__builtin_amdgcn_s_wait_tensorcnt)
#define USE_TDM 1
#else
#define USE_TDM 0
#endif

// LDS layout (dynamic shared memory, assumed to start at LDS offset 0:
// no static __shared__ objects in this TU):
//   [0 .. TS*G4)            : xi tile, packed [s][unit n][gate i,f,g,o]
//   [TS*G4 .. TS*G4 + 2*TS) : x double buffer (2 x TS floats)
#define XI_OFF_FLOATS   0
#define XBUF_OFF_FLOATS (TS * G4)
#define SMEM_FLOATS     (TS * G4 + 2 * TS)

#if __has_builtin(__builtin_amdgcn_tanhf)
__device__ __forceinline__ float fast_tanh(float x) {
    return __builtin_amdgcn_tanhf(x);          // v_tanh_f32 (gfx1250)
}
__device__ __forceinline__ float fast_sigmoid(float x) {
    // sigmoid(x) = 0.5*tanh(x/2) + 0.5 : mul + trans + fma
    return fmaf(0.5f, __builtin_amdgcn_tanhf(0.5f * x), 0.5f);
}
#else
__device__ __forceinline__ float fast_tanh(float x) {
    return 1.0f - 2.0f / (1.0f + __expf(2.0f * x));
}
__device__ __forceinline__ float fast_sigmoid(float x) {
    return 1.0f / (1.0f + __expf(-x));
}
#endif

// Issue an async TDM gather of x[t0 .. t0+TS) = src[t0.. , 0, 0] (stride
// N_FEAT floats) into LDS at byte offset lds_byte_off. Compacts the strided
// column into TS contiguous floats.
__device__ __forceinline__ void load_x_tile(const float* __restrict__ src,
                                            float* __restrict__ xbuf_generic,
                                            unsigned lds_byte_off, int t0) {
#if USE_TDM
    unsigned long long ga =
        (unsigned long long)(uintptr_t)(const void*)(src + (size_t)t0 * N_FEAT);
    v4u g0;
    g0.x = 1u;                                   // count=1, is_restore=0
    g0.y = lds_byte_off;                         // LDS dest byte address
    g0.z = (unsigned)(ga & 0xFFFFFFFFull);       // global_addr[31:0]
    g0.w = (unsigned)((ga >> 32) & 0x1FFFFFFull) // global_addr[56:32]
           | (2u << 30);                         // type=2 ("image")
    v8i g1;
    g1[0] = 0x00020000;          // workgroup_mask=0, data_size=2 (4 bytes)
    g1[1] = (int)(N_FEAT << 16); // tensor_dim0 = 3 (row length, for OOB)
    g1[2] = (int)(SEQ_LEN << 16);// tensor_dim0[31:16]=0 | tensor_dim1=16384
    g1[3] = (int)(1 << 16);      // tensor_dim1[31:16]=0 | tile_dim0=1
    g1[4] = TS;                  // tile_dim1=128 rows, tile_dim2=0
    g1[5] = N_FEAT;              // tensor_dim0_stride = 3 elements
    g1[6] = 0;                   // stride0[47:32]=0, tensor_dim1_stride lo=0
    g1[7] = 0;
    v4i z4 = {0, 0, 0, 0};       // groups 2/3 unused (<=2D tensor)
    v8i z8 = {0, 0, 0, 0, 0, 0, 0, 0};
    __builtin_amdgcn_tensor_load_to_lds(g0, g1, z4, z4, z8, 0);
#else
    (void)lds_byte_off;
    for (int s = (int)threadIdx.x; s < TS; s += 32)
        xbuf_generic[s] = src[(size_t)(t0 + s) * N_FEAT];
#endif
}

__device__ __forceinline__ void wait_x_tile() {
#if USE_TDM
    __builtin_amdgcn_s_wait_tensorcnt(0);
#endif
    __asm__ __volatile__("" ::: "memory");
}

__device__ __forceinline__ float lane_bcast(float v, int k) {
    return __int_as_float(__builtin_amdgcn_readlane(__float_as_int(v), k));
}

__global__ void __launch_bounds__(32, 1)
lstm_scan_kernel(const float* __restrict__ src,
                 const float* __restrict__ W_ih,
                 const float* __restrict__ W_hh,
                 const float* __restrict__ b_ih,
                 const float* __restrict__ b_hh,
                 const float* __restrict__ W_out,
                 const float* __restrict__ b_out,
                 float* __restrict__ out) {
    extern __shared__ float smem[];
    float* xi   = smem + XI_OFF_FLOATS;   // [TS][21][4] packed per unit
    float* xbuf = smem + XBUF_OFF_FLOATS; // [2][TS]

    const int lane = (int)threadIdx.x;

    // ---- per-lane constants for the xi tile compute (rows j, j+32, j+64).
    // Packed position of gate-row j inside a step: (j%21)*4 + (j/21).
    const int j0 = lane, j1 = lane + 32, j2 = lane + 64;
    const int p0 = (j0 % HID) * 4 + (j0 / HID);
    const int p1 = (j1 % HID) * 4 + (j1 / HID);
    const int p2 = (j2 < G4) ? ((j2 % HID) * 4 + (j2 / HID)) : 0;
    const float wih0 = W_ih[j0], bb0 = b_ih[j0] + b_hh[j0];
    const float wih1 = W_ih[j1], bb1 = b_ih[j1] + b_hh[j1];
    float wih2 = 0.0f, bb2 = 0.0f;
    if (j2 < G4) { wih2 = W_ih[j2]; bb2 = b_ih[j2] + b_hh[j2]; }

    // ---- per-lane recurrent weights: lane n owns hidden unit n (n<21)
    const int n = (lane < HID) ? lane : (HID - 1);
    float wi[HID], wf[HID], wg[HID], wo[HID], wq[HID];
#pragma unroll
    for (int k = 0; k < HID; ++k) {
        wi[k] = W_hh[(0 * HID + n) * HID + k];
        wf[k] = W_hh[(1 * HID + n) * HID + k];
        wg[k] = W_hh[(2 * HID + n) * HID + k];
        wo[k] = W_hh[(3 * HID + n) * HID + k];
        wq[k] = W_out[k];
    }
    const float bo = b_out[0];

    // ---- kick off DMA of the first x tile into buffer 0
    load_x_tile(src, xbuf, (unsigned)(XBUF_OFF_FLOATS * 4), 0);

    float h = 0.0f, c = 0.0f;

    for (int tile = 0; tile < NTILES; ++tile) {
        const int buf = tile & 1;
        float* xcur = xbuf + buf * TS;

        wait_x_tile();       // x tile `tile` resident in LDS
        __syncthreads();     // single wave: orders LDS producer/consumer

        // Prefetch next x tile into the other buffer while we compute.
        if (tile + 1 < NTILES) {
            const int nbuf = (tile + 1) & 1;
            load_x_tile(src, xbuf + nbuf * TS,
                        (unsigned)((XBUF_OFF_FLOATS + nbuf * TS) * 4),
                        (tile + 1) * TS);
        }

        // xi[s][n][g] = x[s]*W_ih[j] + (b_ih[j]+b_hh[j]), packed per unit.
        for (int s = 0; s < TS; ++s) {
            const float xs = xcur[s];                // LDS broadcast
            xi[s * G4 + p0] = fmaf(xs, wih0, bb0);
            xi[s * G4 + p1] = fmaf(xs, wih1, bb1);
            if (j2 < G4) xi[s * G4 + p2] = fmaf(xs, wih2, bb2);
        }
        __syncthreads();

        // ---- serial recurrence over this tile
        for (int s = 0; s < TS; ++s) {
            // One ds_load_b128: {i,f,g,o} biases for unit n at step s.
            const float4 q = *(const float4*)(xi + s * G4 + n * 4);
            // Split accumulators: halve the FMA dependency depth.
            float ai0 = q.x, af0 = q.y, ag0 = q.z, ao0 = q.w, oo0 = bo;
            float ai1 = 0.f, af1 = 0.f, ag1 = 0.f, ao1 = 0.f, oo1 = 0.f;
#pragma unroll
            for (int k = 0; k < 11; ++k) {
                const float hk = lane_bcast(h, k);
                ai0 = fmaf(wi[k], hk, ai0);
                af0 = fmaf(wf[k], hk, af0);
                ag0 = fmaf(wg[k], hk, ag0);
                ao0 = fmaf(wo[k], hk, ao0);
                oo0 = fmaf(wq[k], hk, oo0);
            }
#pragma unroll
            for (int k = 11; k < HID; ++k) {
                const float hk = lane_bcast(h, k);
                ai1 = fmaf(wi[k], hk, ai1);
                af1 = fmaf(wf[k], hk, af1);
                ag1 = fmaf(wg[k], hk, ag1);
                ao1 = fmaf(wo[k], hk, ao1);
                oo1 = fmaf(wq[k], hk, oo1);
            }
            const float ai = ai0 + ai1, af = af0 + af1;
            const float ag = ag0 + ag1, ao = ao0 + ao1;
            const float oacc = oo0 + oo1;  // W_out . h_{t-1} + b_out

            const int t = tile * TS + s;
            if (t > 0 && lane == 0) out[t - 1] = oacc;

            const float gi = fast_sigmoid(ai);
            const float gf = fast_sigmoid(af);
            const float gg = fast_tanh(ag);
            const float go = fast_sigmoid(ao);
            c = fmaf(gf, c, gi * gg);
            h = go * fast_tanh(c);
        }
    }

    // ---- out[SEQ_LEN-1] from the final h
    float oacc = bo;
#pragma unroll
    for (int k = 0; k < HID; ++k) {
        oacc = fmaf(wq[k], lane_bcast(h, k), oacc);
    }
    if (lane == 0) out[SEQ_LEN - 1] = oacc;
}

extern "C" void kernel_launch(void* const* d_in, const int* in_sizes, int n_in,
                              void* d_out, int out_size, void* d_ws,
                              size_t ws_size, hipStream_t stream) {
    (void)in_sizes; (void)n_in; (void)out_size; (void)d_ws; (void)ws_size;
    const float* src   = (const float*)d_in[0];
    const float* W_ih  = (const float*)d_in[1];
    const float* W_hh  = (const float*)d_in[2];
    const float* b_ih  = (const float*)d_in[3];
    const float* b_hh  = (const float*)d_in[4];
    const float* W_out = (const float*)d_in[5];
    const float* b_out = (const float*)d_in[6];
    float* out = (float*)d_out;

    const size_t smem_bytes = SMEM_FLOATS * sizeof(float);
    lstm_scan_kernel<<<dim3(1), dim3(32), smem_bytes, stream>>>(
        src, W_ih, W_hh, b_ih, b_hh, W_out, b_out, out);
}